// MyLSTM_90598040142160
// MI455X (gfx1250) — compile-verified
//
#include <hip/hip_runtime.h>
#include <math.h>

typedef float v2f __attribute__((ext_vector_type(2)));
typedef float v8f __attribute__((ext_vector_type(8)));

#define DIM  2048   // input dim d
#define HID  2048   // hidden dim h
#define KCH  1024   // K-chunk per block (4 chunks cover concat(x,h0) = 4096)
#define TK   32     // K-rows per staged tile
#define TN   128    // columns per block
#define NT   (KCH / TK)          // 32 tiles per block
#define TILE_FLOATS (TK * TN)    // 4096 floats = 16 KB

// ---------------------------------------------------------------------------
// Kernel A: partial GEMV via V_WMMA_F32_16X16X4_F32 with double-buffered
// GLOBAL_LOAD_ASYNC_TO_LDS_B128 weight staging (ASYNCcnt pipeline).
// grid = 4 gates x 16 col-blocks x 4 k-chunks = 256 blocks, 256 thr (8 waves).
// partial layout: float[4 kchunks][4 gates][HID]
// ---------------------------------------------------------------------------
__global__ __launch_bounds__(256)
void lstm_gemv_wmma(const float* __restrict__ x,  const float* __restrict__ h0,
                    const float* __restrict__ wxi, const float* __restrict__ wxf,
                    const float* __restrict__ wxo, const float* __restrict__ wxc,
                    const float* __restrict__ whi, const float* __restrict__ whf,
                    const float* __restrict__ who, const float* __restrict__ whc,
                    float* __restrict__ partial)
{
    __shared__ float xs[KCH];                 // 4 KB vector stage
    __shared__ float tile[2][TILE_FLOATS];    // 2 x 16 KB weight tiles

    const int tid  = threadIdx.x;
    const int bx   = blockIdx.x;
    const int kc   = bx & 3;           // k-chunk 0..3
    const int colb = (bx >> 2) & 15;   // column block 0..15 (128 cols each)
    const int g    = bx >> 6;          // gate 0..3 (i,f,o,c)

    // Stage the 1024-float vector chunk into LDS (one float4 per thread).
    const float* vsrc = (kc < 2) ? (x + kc * KCH) : (h0 + (kc - 2) * KCH);
    *(float4*)&xs[tid * 4] = *(const float4*)&vsrc[tid * 4];

    // Weight matrix base for this (gate, k-chunk), plus this block's column base.
    const float* W;
    if (kc < 2)
        W = (g == 0 ? wxi : g == 1 ? wxf : g == 2 ? wxo : wxc) + (size_t)kc * KCH * HID;
    else
        W = (g == 0 ? whi : g == 1 ? whf : g == 2 ? who : whc) + (size_t)(kc - 2) * KCH * HID;
    const unsigned long long wbase = (unsigned long long)(uintptr_t)W;
    const unsigned colbase_bytes   = (unsigned)(colb * TN) * 4u;

    // LDS byte offsets of the two tile buffers (generic addr low 32 bits = LDS offset).
    const unsigned lds_t0 = (unsigned)(uintptr_t)&tile[0][0];
    const unsigned lds_t1 = (unsigned)(uintptr_t)&tile[1][0];

    // This thread's slice of a tile: 4 x b128 chunks (row = li/32, 16B col chunk).
    // Precompute per-q row/col from linear index li = q*256 + tid.
    const int wave = tid >> 5;
    const int lane = tid & 31;
    const int nloc = wave * 16;                // wave's 16 columns within the tile
    const int cofs = (lane >> 4) * 2;          // K sub-offset: 0 / 2 per lane half
    const int nl   = lane & 15;
    const bool arow = (nl == 0);               // lanes 0 & 16 carry A row M=0

    v8f acc = {0.f, 0.f, 0.f, 0.f, 0.f, 0.f, 0.f, 0.f};

    // Issue one tile's async loads: 4 instructions/wave -> ASYNCcnt += 4.
    #define ISSUE_TILE(T, LDSBASE)                                              \
        do {                                                                    \
            const int _r0 = (T) * TK;                                           \
            _Pragma("unroll")                                                   \
            for (int _q = 0; _q < 4; ++_q) {                                    \
                const int _li   = _q * 256 + tid;                               \
                const int _row  = _li >> 5;          /* 0..31 */                \
                const int _cid  = (_li & 31) << 2;   /* float col 0..124 */     \
                const unsigned _goff = (unsigned)((_r0 + _row) * HID + _cid) * 4u \
                                       + colbase_bytes;                         \
                const unsigned _loff = (LDSBASE) + (unsigned)(_row * TN + _cid) * 4u; \
                asm volatile("global_load_async_to_lds_b128 %0, %1, %2"         \
                             :: "v"(_loff), "v"(_goff), "s"(wbase) : "memory"); \
            }                                                                   \
        } while (0)

    ISSUE_TILE(0, lds_t0);

    for (int t = 0; t < NT; ++t) {
        const unsigned curbase = (t & 1) ? lds_t1 : lds_t0;

        if (t + 1 < NT) {
            ISSUE_TILE(t + 1, (t & 1) ? lds_t0 : lds_t1);
            // keep newest 4 (tile t+1) in flight; tile t is complete
            asm volatile("s_wait_asynccnt 0x4" ::: "memory");
        } else {
            asm volatile("s_wait_asynccnt 0x0" ::: "memory");
        }
        __syncthreads();   // tile t visible to all waves (and xs on t==0)

        const float* tb = (const float*)(&tile[t & 1][0]);
        const int r0 = t * TK;
        #pragma unroll
        for (int kk = 0; kk < TK; kk += 4) {
            // A fragment: vector values only in matrix row M=0 (lanes 0 & 16).
            float2 t2 = *(const float2*)&xs[r0 + kk + cofs];   // ds_load_b64
            v2f a;
            a[0] = arow ? t2.x : 0.0f;
            a[1] = arow ? t2.y : 0.0f;

            // B fragment from the staged LDS tile.
            v2f b;
            b[0] = tb[(kk + cofs)     * TN + nloc + nl];
            b[1] = tb[(kk + cofs + 1) * TN + nloc + nl];

            acc = __builtin_amdgcn_wmma_f32_16x16x4_f32(
                      /*neg_a=*/false, a, /*neg_b=*/false, b,
                      /*c_mod=*/(short)0, acc, /*reuse_a=*/false, /*reuse_b=*/false);
        }
        __syncthreads();   // done reading buffer before it is refilled
    }
    #undef ISSUE_TILE

    // Row M=0 of C sits in C-VGPR0 on lanes 0..15 (N = lane) -> acc[0].
    if (lane < 16)
        partial[((size_t)kc * 4 + g) * HID + colb * TN + nloc + lane] = acc[0];
}

// ---------------------------------------------------------------------------
// Kernel B: sum the 4 K-chunk partials, add bias, apply sigmoid.
// gates layout: float[4][HID]  (i, f, o, c)
// ---------------------------------------------------------------------------
__global__ __launch_bounds__(256)
void lstm_activate(const float* __restrict__ partial,
                   const float* __restrict__ bi, const float* __restrict__ bf,
                   const float* __restrict__ bo, const float* __restrict__ bcv,
                   float* __restrict__ gates)
{
    const int idx = blockIdx.x * 256 + threadIdx.x;   // 0 .. 4*HID-1
    if (idx >= 4 * HID) return;
    const int g = idx >> 11;          // / HID
    const int n = idx & (HID - 1);

    float v = partial[((size_t)0 * 4 + g) * HID + n]
            + partial[((size_t)1 * 4 + g) * HID + n]
            + partial[((size_t)2 * 4 + g) * HID + n]
            + partial[((size_t)3 * 4 + g) * HID + n];
    v += (g == 0 ? bi : g == 1 ? bf : g == 2 ? bo : bcv)[n];
    gates[idx] = 1.0f / (1.0f + expf(-v));
}

// ---------------------------------------------------------------------------
// Kernel C: tail. s = dot(i,c); cell = f*c0 + s; normalize (ddof=1); h=o*tanh.
// Single block, 256 threads, 8 elements each, LDS tree reductions.
// ---------------------------------------------------------------------------
__global__ __launch_bounds__(256)
void lstm_tail(const float* __restrict__ gates,
               const float* __restrict__ c0v,
               float* __restrict__ hout)
{
    __shared__ float red[256];
    __shared__ float bcast[3];

    const int tid = threadIdx.x;
    const float* ig = gates;
    const float* fg = gates + HID;
    const float* og = gates + 2 * HID;
    const float* cg = gates + 3 * HID;

    // 1) s = dot(i, c)
    float acc = 0.f;
    for (int j = tid; j < HID; j += 256) acc += ig[j] * cg[j];
    red[tid] = acc; __syncthreads();
    for (int off = 128; off > 0; off >>= 1) {
        if (tid < off) red[tid] += red[tid + off];
        __syncthreads();
    }
    if (tid == 0) bcast[0] = red[0];
    __syncthreads();
    const float s = bcast[0];

    // 2) cell = f*c0 + s ; mean
    float cell[HID / 256];
    float sum = 0.f;
    for (int r = 0; r < HID / 256; ++r) {
        const int j = tid + r * 256;
        cell[r] = fg[j] * c0v[j] + s;
        sum += cell[r];
    }
    red[tid] = sum; __syncthreads();
    for (int off = 128; off > 0; off >>= 1) {
        if (tid < off) red[tid] += red[tid + off];
        __syncthreads();
    }
    if (tid == 0) bcast[1] = red[0] / (float)HID;
    __syncthreads();
    const float mean = bcast[1];

    // 3) std (unbiased, ddof=1)
    float var = 0.f;
    for (int r = 0; r < HID / 256; ++r) {
        const float d = cell[r] - mean;
        var += d * d;
    }
    red[tid] = var; __syncthreads();
    for (int off = 128; off > 0; off >>= 1) {
        if (tid < off) red[tid] += red[tid + off];
        __syncthreads();
    }
    if (tid == 0) bcast[2] = sqrtf(red[0] / (float)(HID - 1));
    __syncthreads();
    const float stdv = bcast[2];

    // 4) h = o * tanh((cell - mean) / (std + 1e-5))
    for (int r = 0; r < HID / 256; ++r) {
        const int j = tid + r * 256;
        hout[j] = og[j] * tanhf((cell[r] - mean) / (stdv + 1e-5f));
    }
}

// ---------------------------------------------------------------------------
// Host-side launcher (graph-capture safe: only kernel launches on `stream`).
// ---------------------------------------------------------------------------
extern "C" void kernel_launch(void* const* d_in, const int* in_sizes, int n_in,
                              void* d_out, int out_size, void* d_ws, size_t ws_size,
                              hipStream_t stream)
{
    const float* x   = (const float*)d_in[0];
    const float* wxi = (const float*)d_in[1];
    const float* wxf = (const float*)d_in[2];
    const float* wxo = (const float*)d_in[3];
    const float* wxc = (const float*)d_in[4];
    const float* whi = (const float*)d_in[5];
    const float* whf = (const float*)d_in[6];
    const float* who = (const float*)d_in[7];
    const float* whc = (const float*)d_in[8];
    const float* bi  = (const float*)d_in[9];
    const float* bf  = (const float*)d_in[10];
    const float* bo  = (const float*)d_in[11];
    const float* bcv = (const float*)d_in[12];
    const float* h0  = (const float*)d_in[13];
    const float* c0  = (const float*)d_in[14];

    float* partial = (float*)d_ws;                 // 4*4*2048 floats = 128 KB
    float* gates   = partial + 4 * 4 * HID;        // 4*2048 floats  =  32 KB

    lstm_gemv_wmma<<<256, 256, 0, stream>>>(x, h0,
                                            wxi, wxf, wxo, wxc,
                                            whi, whf, who, whc,
                                            partial);
    lstm_activate<<<(4 * HID + 255) / 256, 256, 0, stream>>>(partial, bi, bf, bo, bcv, gates);
    lstm_tail<<<1, 256, 0, stream>>>(gates, c0, (float*)d_out);
}